// RelationalAttention_33827162423518
// MI455X (gfx1250) — compile-verified
//
#include <hip/hip_runtime.h>
#include <math.h>

typedef __attribute__((ext_vector_type(2))) float v2f;
typedef __attribute__((ext_vector_type(8))) float v8f;

#define N_NODES   100000
#define N_EDGES   625000
#define NODE_DIM  128
#define N_REL     16
#define SLOPE     0.2f
#define NTILES    (N_NODES / 16)        // 6250 exact
#define WPB       4                     // waves (16-node tiles) per block

// ---------------------------------------------------------------------------
// Kernel 1: S[n][0..15]  = sum_k a_r[r][k]     * x[n][k]   (src scores)
//           S[n][16..31] = sum_k a_r[r][128+k] * x[n][k]   (dst scores)
// Skinny GEMM X[100000x128] @ W[128x32] via V_WMMA_F32_16X16X4_F32.
//
// LDS holds W pair-interleaved: Wp[p][n][j] = W[2p+j][n]  (p = k/2, j = 0..1)
// so a lane's B fragment {W[kk][n], W[kk+1][n]} is one aligned ds_load_b64.
// ---------------------------------------------------------------------------
__global__ __launch_bounds__(WPB * 32) void precompute_scores_kernel(
    const float* __restrict__ x,      // [N_NODES, 128]
    const float* __restrict__ a_r,    // [16, 256]
    float* __restrict__ S)            // [N_NODES, 32]
{
    __shared__ float Wlds[NODE_DIM * 32];   // 16 KB, layout [k/2][n][2]
    for (int idx = threadIdx.x; idx < NODE_DIM * 32; idx += WPB * 32) {
        int p   = idx >> 6;          // k-pair index 0..63
        int rem = idx & 63;
        int n   = rem >> 1;          // column 0..31
        int k   = (p << 1) | (rem & 1);
        // W[k][n]: n<16 -> a_r[n][k]; n>=16 -> a_r[n-16][128+k]
        Wlds[idx] = a_r[(n & 15) * (2 * NODE_DIM) + ((n >> 4) << 7) + k];
    }
    __syncthreads();

    int lane = threadIdx.x & 31;
    int wave = threadIdx.x >> 5;
    int tile = blockIdx.x * WPB + wave;
    if (tile >= NTILES) return;           // wave-uniform: EXEC stays all-1s

    int m     = lane & 15;                // matrix row within tile
    int nn    = lane & 15;                // B/C/D column index
    int khalf = (lane >> 4) << 1;         // 0 for lanes 0-15, 2 for lanes 16-31

    const float* __restrict__ xrow = x + (size_t)(tile * 16 + m) * NODE_DIM;

    v8f acc_s = {};                       // N = 0..15  (src scores)
    v8f acc_d = {};                       // N = 16..31 (dst scores)

    #pragma unroll
    for (int kb = 0; kb < NODE_DIM; kb += 4) {
        int kk = kb + khalf;              // even
        // A fragment (16x4 f32): lane m, VGPR0 = K kk, VGPR1 = K kk+1
        v2f a = *(const v2f*)&xrow[kk];   // global_load_b64
        // B fragments: one aligned ds_load_b64 each
        const int pbase = (kk >> 1) * 64;
        v2f bs = *(const v2f*)&Wlds[pbase + nn * 2];         // cols 0..15
        v2f bd = *(const v2f*)&Wlds[pbase + (16 + nn) * 2];  // cols 16..31

        acc_s = __builtin_amdgcn_wmma_f32_16x16x4_f32(
            false, a, false, bs, (short)0, acc_s, false, false);
        acc_d = __builtin_amdgcn_wmma_f32_16x16x4_f32(
            false, a, false, bd, (short)0, acc_d, false, false);
    }

    // C/D layout: VGPR v holds M = v + 8*(lane>=16), N = lane&15
    int mtop = (lane >> 4) * 8;
    size_t rowbase = (size_t)(tile * 16 + mtop) * 32;
    #pragma unroll
    for (int v = 0; v < 8; ++v) {
        S[rowbase + (size_t)v * 32 + nn]      = acc_s[v];
        S[rowbase + (size_t)v * 32 + 16 + nn] = acc_d[v];
    }
}

// ---------------------------------------------------------------------------
// Kernel 0: init segment max/sum
// ---------------------------------------------------------------------------
__global__ void init_segments_kernel(float* __restrict__ seg_max,
                                     float* __restrict__ seg_sum)
{
    int i = blockIdx.x * blockDim.x + threadIdx.x;
    if (i < N_NODES) {
        seg_max[i] = -__builtin_huge_valf();
        seg_sum[i] = 0.0f;
    }
}

// ---------------------------------------------------------------------------
// Kernel 2: per-edge logit + leaky relu, atomic segment max over src
// ---------------------------------------------------------------------------
__global__ void edge_logits_kernel(const float* __restrict__ S,
                                   const int* __restrict__ eidx,   // [2, E]
                                   const int* __restrict__ etype,  // [E]
                                   float* __restrict__ e_act,      // [E] (d_out)
                                   float* __restrict__ seg_max)
{
    int i = blockIdx.x * blockDim.x + threadIdx.x;
    if (i >= N_EDGES) return;
    int src = eidx[i];
    int dst = eidx[N_EDGES + i];
    int r   = etype[i];
    float e = S[(size_t)src * 32 + r] + S[(size_t)dst * 32 + 16 + r];
    e = (e >= 0.0f) ? e : SLOPE * e;
    e_act[i] = e;
    atomicMax(&seg_max[src], e);          // global_atomic_max_num_f32
}

// ---------------------------------------------------------------------------
// Kernel 3: ex = exp(e - segmax[src]); atomic segment sum
// ---------------------------------------------------------------------------
__global__ void edge_exp_kernel(const int* __restrict__ eidx,
                                const float* __restrict__ seg_max,
                                float* __restrict__ ex,             // in/out (d_out)
                                float* __restrict__ seg_sum)
{
    int i = blockIdx.x * blockDim.x + threadIdx.x;
    if (i >= N_EDGES) return;
    int src = eidx[i];
    float v = __expf(ex[i] - seg_max[src]);
    ex[i] = v;
    atomicAdd(&seg_sum[src], v);          // global_atomic_add_f32
}

// ---------------------------------------------------------------------------
// Kernel 4: alpha = ex / segsum[src]
// ---------------------------------------------------------------------------
__global__ void edge_norm_kernel(const int* __restrict__ eidx,
                                 const float* __restrict__ seg_sum,
                                 float* __restrict__ alpha)          // d_out
{
    int i = blockIdx.x * blockDim.x + threadIdx.x;
    if (i >= N_EDGES) return;
    int src = eidx[i];
    alpha[i] = alpha[i] / seg_sum[src];
}

// ---------------------------------------------------------------------------
extern "C" void kernel_launch(void* const* d_in, const int* in_sizes, int n_in,
                              void* d_out, int out_size, void* d_ws, size_t ws_size,
                              hipStream_t stream)
{
    const float* x     = (const float*)d_in[0];   // [100000,128]
    const float* a_r   = (const float*)d_in[1];   // [16,256]
    const int*   eidx  = (const int*)d_in[2];     // [2,625000]
    const int*   etype = (const int*)d_in[3];     // [625000]
    float*       out   = (float*)d_out;           // [625000] alpha

    float* S       = (float*)d_ws;                         // 100000*32 f32
    float* seg_max = S + (size_t)N_NODES * 32;             // 100000 f32
    float* seg_sum = seg_max + N_NODES;                    // 100000 f32

    // init segment accumulators
    init_segments_kernel<<<(N_NODES + 255) / 256, 256, 0, stream>>>(seg_max, seg_sum);

    // WMMA skinny GEMM: per-node, per-relation scores
    precompute_scores_kernel<<<(NTILES + WPB - 1) / WPB, WPB * 32, 0, stream>>>(
        x, a_r, S);

    const int EB = (N_EDGES + 255) / 256;
    edge_logits_kernel<<<EB, 256, 0, stream>>>(S, eidx, etype, out, seg_max);
    edge_exp_kernel<<<EB, 256, 0, stream>>>(eidx, seg_max, out, seg_sum);
    edge_norm_kernel<<<EB, 256, 0, stream>>>(eidx, seg_sum, out);
}